// TemplatePredictionHead_6459630814081
// MI455X (gfx1250) — compile-verified
//
#include <hip/hip_runtime.h>
#include <hip/hip_bf16.h>

typedef float v2f __attribute__((ext_vector_type(2)));
typedef float v8f __attribute__((ext_vector_type(8)));

#define D_DIM   768
#define N_ATOM  1001
#define N_BOND  301
#define M_TOK   2048   // 8 * 256

// ---------------------------------------------------------------------------
// GEMM: out[m][n] = sum_k X[m*K + k] * W[n*ldw + k]  (+ bias[n] if bias)
// One wave (32 threads) computes a 32x64 output tile with
// V_WMMA_F32_16X16X4_F32: 2 M-subtiles x 4 N-subtiles = 8 accumulators.
// K loop is software-pipelined: fragments for step k+4 are loaded before
// the WMMAs consuming step k's fragments, so loads overlap WMMA latency
// instead of serializing on s_wait_loadcnt 0.
//
// Fragment layout per ISA 7.12.2 (32-bit A 16x4, 32-bit C/D 16x16), wave32:
//   lane = 16*h + idx  (h = lane>>4, idx = lane&15)
//   A: VGPR0/1 = A[idx][k+2h], A[idx][k+2h+1]
//   B: VGPR0/1 = B[k+2h][n0+idx], B[k+2h+1][n0+idx]  (B = W^T -> rows of W)
//   D: VGPR r  = D[r + 8h][idx]
// ---------------------------------------------------------------------------
__global__ __launch_bounds__(32)
void gemm_xwT_wmma(const float* __restrict__ X,
                   const float* __restrict__ W,
                   const float* __restrict__ bias,
                   float* __restrict__ out,
                   int N, int K, int ldw, int ldo)
{
    const int lane = threadIdx.x & 31;
    const int half = lane >> 4;
    const int idx  = lane & 15;
    const int m0   = blockIdx.x * 32;
    const int n0   = blockIdx.y * 64;

    const float* xrow0 = X + (size_t)(m0 + idx) * K;        // A rows, subtile 0
    const float* xrow1 = X + (size_t)(m0 + 16 + idx) * K;   // A rows, subtile 1

    int          ncol[4];
    bool         nok[4];
    const float* wrow[4];
    v8f          acc[2][4];
#pragma unroll
    for (int t = 0; t < 4; ++t) {
        ncol[t] = n0 + t * 16 + idx;
        nok[t]  = ncol[t] < N;
        // clamp OOB columns to row 0 (valid memory); results are discarded.
        wrow[t]    = W + (size_t)(nok[t] ? ncol[t] : 0) * ldw;
        acc[0][t]  = (v8f)(0.0f);
        acc[1][t]  = (v8f)(0.0f);
    }

    const int koff = 2 * half;

    // ---- prologue: load fragments for k = 0 ----
    v2f a0 = *(const v2f*)(xrow0 + koff);
    v2f a1 = *(const v2f*)(xrow1 + koff);
    v2f bf[4];
#pragma unroll
    for (int t = 0; t < 4; ++t) bf[t] = *(const v2f*)(wrow[t] + koff);

    // ---- pipelined main loop: prefetch k+4, then consume k ----
#pragma unroll 2
    for (int k = 0; k + 4 < K; k += 4) {
        const int kn = k + 4 + koff;
        v2f a0n = *(const v2f*)(xrow0 + kn);
        v2f a1n = *(const v2f*)(xrow1 + kn);
        v2f bfn[4];
#pragma unroll
        for (int t = 0; t < 4; ++t) bfn[t] = *(const v2f*)(wrow[t] + kn);

#pragma unroll
        for (int t = 0; t < 4; ++t)
            acc[0][t] = __builtin_amdgcn_wmma_f32_16x16x4_f32(
                false, a0, false, bf[t], (short)0, acc[0][t], false, false);
#pragma unroll
        for (int t = 0; t < 4; ++t)
            acc[1][t] = __builtin_amdgcn_wmma_f32_16x16x4_f32(
                false, a1, false, bf[t], (short)0, acc[1][t], false, false);

        a0 = a0n;
        a1 = a1n;
#pragma unroll
        for (int t = 0; t < 4; ++t) bf[t] = bfn[t];
    }

    // ---- epilogue: consume final fragments (k = K-4) ----
#pragma unroll
    for (int t = 0; t < 4; ++t)
        acc[0][t] = __builtin_amdgcn_wmma_f32_16x16x4_f32(
            false, a0, false, bf[t], (short)0, acc[0][t], false, false);
#pragma unroll
    for (int t = 0; t < 4; ++t)
        acc[1][t] = __builtin_amdgcn_wmma_f32_16x16x4_f32(
            false, a1, false, bf[t], (short)0, acc[1][t], false, false);

    // ---- store: element (m = r + 8h, n = n0+16t+idx) lives in acc[s][t][r] ----
#pragma unroll
    for (int t = 0; t < 4; ++t) {
        if (!nok[t]) continue;
        const float bv = bias ? bias[ncol[t]] : 0.0f;
        float* ocol = out + ncol[t];
#pragma unroll
        for (int s = 0; s < 2; ++s) {
#pragma unroll
            for (int r = 0; r < 8; ++r) {
                const int m = m0 + 16 * s + r + 8 * half;
                ocol[(size_t)m * ldo] = acc[s][t][r] + bv;
            }
        }
    }
}

// ---------------------------------------------------------------------------
// bond_logits[b,i,j,n] = p1[b*256+i][n] + p2[b*256+j][n] + bb[n]
// One block per (b,i). p1 row + bb folded into LDS once; loop j streaming
// p2 rows (L2-resident, 2.4 MB) and emitting nontemporal stores (631 MB
// write-once output > 192 MB L2, so keep it out of cache).
// ---------------------------------------------------------------------------
__global__ __launch_bounds__(320)
void bond_expand(const float* __restrict__ p1,
                 const float* __restrict__ p2,
                 const float* __restrict__ bb,
                 float* __restrict__ out)
{
    const int bi  = blockIdx.x;        // b*256 + i
    const int b   = bi >> 8;
    const int tid = threadIdx.x;

    __shared__ float p1s[N_BOND];
    if (tid < N_BOND) p1s[tid] = p1[(size_t)bi * N_BOND + tid] + bb[tid];
    __syncthreads();

    float*       orow   = out + (size_t)bi * (256 * N_BOND);
    const float* p2base = p2 + (size_t)b * 256 * N_BOND;

    if (tid < N_BOND) {
        const float pv = p1s[tid];
#pragma unroll 4
        for (int j = 0; j < 256; ++j) {
            const float v = pv + p2base[(size_t)j * N_BOND + tid];
            __builtin_nontemporal_store(v, orow + (size_t)j * N_BOND + tid);
        }
    }
}

// ---------------------------------------------------------------------------
extern "C" void kernel_launch(void* const* d_in, const int* in_sizes, int n_in,
                              void* d_out, int out_size, void* d_ws, size_t ws_size,
                              hipStream_t stream)
{
    const float* x  = (const float*)d_in[0];  // (8,256,768)
    const float* Wa = (const float*)d_in[1];  // (1001,768)
    const float* ba = (const float*)d_in[2];  // (1001,)
    const float* Wb = (const float*)d_in[3];  // (301,1536)
    const float* bb = (const float*)d_in[4];  // (301,)

    float* atom_out = (float*)d_out;                               // 8*256*1001
    float* bond_out = atom_out + (size_t)M_TOK * N_ATOM;           // 8*256*256*301

    float* p1 = (float*)d_ws;                                      // 2048*301
    float* p2 = p1 + (size_t)M_TOK * N_BOND;                       // 2048*301

    dim3 blk(32);

    // atom_logits = X @ Wa^T + ba
    {
        dim3 grid(M_TOK / 32, (N_ATOM + 63) / 64);
        gemm_xwT_wmma<<<grid, blk, 0, stream>>>(x, Wa, ba, atom_out,
                                                N_ATOM, D_DIM, D_DIM, N_ATOM);
    }
    // p1 = X @ W1^T   (W1 = Wb[:, :768], row stride 1536)
    {
        dim3 grid(M_TOK / 32, (N_BOND + 63) / 64);
        gemm_xwT_wmma<<<grid, blk, 0, stream>>>(x, Wb, nullptr, p1,
                                                N_BOND, D_DIM, 2 * D_DIM, N_BOND);
    }
    // p2 = X @ W2^T   (W2 = Wb[:, 768:], row stride 1536)
    {
        dim3 grid(M_TOK / 32, (N_BOND + 63) / 64);
        gemm_xwT_wmma<<<grid, blk, 0, stream>>>(x, Wb + D_DIM, nullptr, p2,
                                                N_BOND, D_DIM, 2 * D_DIM, N_BOND);
    }
    // bond_logits = p1[:, :, None, :] + p2[:, None, :, :] + bb
    {
        dim3 grid(M_TOK);
        bond_expand<<<grid, dim3(320), 0, stream>>>(p1, p2, bb, bond_out);
    }
}